// NeuralOffsetMinSumDecoder_13640816132466
// MI455X (gfx1250) — compile-verified
//
#include <hip/hip_runtime.h>
#include <stdint.h>

static constexpr int Mdim = 2048;
static constexpr int Ndim = 4096;
static constexpr int WPR  = Ndim / 32;        // mask words per row
static constexpr int ROWS_PER_BLOCK = 4;      // 4 waves/block, 1 row/wave

typedef float f4  __attribute__((ext_vector_type(4)));
typedef float v2f __attribute__((ext_vector_type(2)));
typedef float v8f __attribute__((ext_vector_type(8)));
typedef int   v4i __attribute__((ext_vector_type(4)));

typedef __attribute__((address_space(1))) v4i* g_v4i_ptr;   // global
typedef __attribute__((address_space(3))) v4i* l_v4i_ptr;   // LDS

#if __has_builtin(__builtin_amdgcn_global_load_async_to_lds_b128)
#define USE_ASYNC_LDS 1
#endif
#if __has_builtin(__builtin_amdgcn_wmma_f32_16x16x4_f32)
#define USE_WMMA_F32 1
#endif

__device__ __forceinline__ float fsign(float v) {
    return (v > 0.f) ? 1.f : ((v < 0.f) ? -1.f : 0.f);   // matches jnp.sign (0 -> 0)
}

// ---- pack H (dense float mask) into bits: one ballot word per wave32 ----
__global__ void k_pack_mask(const float* __restrict__ H, uint32_t* __restrict__ bits) {
    size_t tid = (size_t)blockIdx.x * blockDim.x + threadIdx.x;
    bool b = H[tid] > 0.5f;
    unsigned long long m = __ballot(b);                  // wave32: low 32 bits
    if ((threadIdx.x & 31) == 0) bits[tid >> 5] = (uint32_t)m;
}

// ---- v2c0 = mask ? llr : 0 ----
__global__ void k_init_v2c(const uint32_t* __restrict__ bits, const float* __restrict__ llr,
                           float* __restrict__ v2c) {
    size_t tid = (size_t)blockIdx.x * blockDim.x + threadIdx.x;
    size_t e = tid * 4;
    int col = (int)(e & (size_t)(Ndim - 1));
    uint32_t w = bits[e >> 5] >> (col & 31);
    f4 l4 = *(const f4*)(llr + col);
    f4 o;
    o.x = (w & 1u) ? l4.x : 0.f;
    o.y = (w & 2u) ? l4.y : 0.f;
    o.z = (w & 4u) ? l4.z : 0.f;
    o.w = (w & 8u) ? l4.w : 0.f;
    *(f4*)(v2c + e) = o;
}

__global__ void k_zero(float* __restrict__ p, int n) {
    int i = blockIdx.x * blockDim.x + threadIdx.x;
    if (i < n) p[i] = 0.f;
}

// ---- check-node update: one row per wave ----
__global__ void __launch_bounds__(128)
k_check(const float* __restrict__ v2c, const float* __restrict__ beta_t,
        const uint32_t* __restrict__ bits, float* __restrict__ c2v) {
    __shared__ __align__(16) float srow[ROWS_PER_BLOCK][Ndim];   // 64 KB
    const int wave = threadIdx.x >> 5;
    const int lane = threadIdx.x & 31;
    const int row  = blockIdx.x * ROWS_PER_BLOCK + wave;
    const float* __restrict__ vrow = v2c + (size_t)row * Ndim;

    // Stage the v2c row into LDS (read twice below). Async DMA on gfx1250.
#ifdef USE_ASYNC_LDS
    for (int i = 0; i < Ndim / 128; ++i) {
        __builtin_amdgcn_global_load_async_to_lds_b128(
            (g_v4i_ptr)(vrow + i * 128 + lane * 4),
            (l_v4i_ptr)&srow[wave][i * 128 + lane * 4],
            0, 0);
    }
#if __has_builtin(__builtin_amdgcn_s_wait_asynccnt)
    __builtin_amdgcn_s_wait_asynccnt(0);
#else
    asm volatile("s_wait_asynccnt 0" ::: "memory");
#endif
#else
    for (int i = 0; i < Ndim / 128; ++i)
        *(f4*)&srow[wave][i * 128 + lane * 4] = *(const f4*)(vrow + i * 128 + lane * 4);
#endif

    // Pass 1: masked min1/min2/argmin(first) + sign product, sparse bit-iteration.
    float min1 = __builtin_inff(), min2 = __builtin_inff(), sgn = 1.0f;
    int idx = 0x7FFFFFFF;
    const uint32_t* __restrict__ wrow = bits + (size_t)row * WPR;
    for (int i = lane; i < WPR; i += 32) {
        uint32_t w = wrow[i];
        while (w) {
            int j = __builtin_ctz(w);
            w &= w - 1;
            int col = i * 32 + j;
            float v = srow[wave][col];
            float a = fabsf(v);
            sgn *= fsign(v);
            if (a < min1) { min2 = min1; min1 = a; idx = col; }
            else          { min2 = fminf(min2, a); }
        }
    }
    // wave32 two-min merge with first-index tie break (matches jnp.argmin)
    for (int off = 16; off; off >>= 1) {
        float o1 = __shfl_xor(min1, off, 32);
        float o2 = __shfl_xor(min2, off, 32);
        int   oi = __shfl_xor(idx,  off, 32);
        float os = __shfl_xor(sgn,  off, 32);
        sgn *= os;
        if (o1 < min1 || (o1 == min1 && oi < idx)) {
            min2 = fminf(min1, o2); min1 = o1; idx = oi;
        } else {
            min2 = fminf(min2, o1);
        }
    }
    if (min2 == __builtin_inff()) min2 = min1;   // degree-1 row

    // Pass 2: c2v = mask ? sprod*sign(v)*relu(raw - beta) : 0
    const float* __restrict__ brow = beta_t + (size_t)row * Ndim;
    float* __restrict__ crow = c2v + (size_t)row * Ndim;
    for (int i = 0; i < Ndim / 128; ++i) {
        int col = i * 128 + lane * 4;
        uint32_t w = (wrow[col >> 5] >> (col & 31)) & 0xFu;
        f4 b4 = __builtin_nontemporal_load((const f4*)(brow + col));  // beta streamed once
        const float* sv = &srow[wave][col];
        f4 o;
        float r;
        r   = (col + 0 == idx) ? min2 : min1;
        o.x = (w & 1u) ? (sgn * fsign(sv[0]) * fmaxf(r - b4.x, 0.f)) : 0.f;
        r   = (col + 1 == idx) ? min2 : min1;
        o.y = (w & 2u) ? (sgn * fsign(sv[1]) * fmaxf(r - b4.y, 0.f)) : 0.f;
        r   = (col + 2 == idx) ? min2 : min1;
        o.z = (w & 4u) ? (sgn * fsign(sv[2]) * fmaxf(r - b4.z, 0.f)) : 0.f;
        r   = (col + 3 == idx) ? min2 : min1;
        o.w = (w & 8u) ? (sgn * fsign(sv[3]) * fmaxf(r - b4.w, 0.f)) : 0.f;
        *(f4*)(crow + col) = o;
    }
}

// ---- colsum[n] = sum_m c2v[m][n] via ones-A WMMA (fp32, K=4), M split 8 ways ----
__global__ void __launch_bounds__(256)
k_colsum(const float* __restrict__ c2v, float* __restrict__ colsum) {
    const int lane    = threadIdx.x & 31;
    const int wglobal = blockIdx.x * (blockDim.x >> 5) + (threadIdx.x >> 5);
    const int ntile   = wglobal & (Ndim / 16 - 1);     // 256 column tiles
    const int mchunk  = wglobal >> 8;                  // 8 row chunks of 256
    const int col     = ntile * 16 + (lane & 15);
    const int m0      = mchunk * (Mdim / 8);
#ifdef USE_WMMA_F32
    const int krow = lane >> 4;                        // B 4x16 layout: vgpr0 K={0,1}, vgpr1 K={2,3}
    v2f a; a.x = 1.0f; a.y = 1.0f;                     // all-ones A -> D rows = column sums
    v8f c = {0.f, 0.f, 0.f, 0.f, 0.f, 0.f, 0.f, 0.f};
    for (int k = m0; k < m0 + Mdim / 8; k += 4) {
        v2f b;
        b.x = c2v[(size_t)(k + krow)     * Ndim + col];
        b.y = c2v[(size_t)(k + krow + 2) * Ndim + col];
        c = __builtin_amdgcn_wmma_f32_16x16x4_f32(false, a, false, b, (short)0, c,
                                                  false, false);
    }
    if (lane < 16) atomicAdd(colsum + col, c[0]);      // D row M=0 = chunk column sum
#else
    float s = 0.f;
    for (int k = m0; k < m0 + Mdim / 8; ++k) s += c2v[(size_t)k * Ndim + col];
    if (lane < 16) atomicAdd(colsum + col, s);
#endif
}

// ---- variable-node update ----
__global__ void k_var(const float* __restrict__ c2v, const uint32_t* __restrict__ bits,
                      const float* __restrict__ llr, const float* __restrict__ colsum,
                      float* __restrict__ v2c) {
    size_t tid = (size_t)blockIdx.x * blockDim.x + threadIdx.x;
    size_t e = tid * 4;
    int col = (int)(e & (size_t)(Ndim - 1));
    uint32_t w = bits[e >> 5] >> (col & 31);
    f4 cv = *(const f4*)(c2v + e);
    f4 l4 = *(const f4*)(llr + col);
    f4 s4 = *(const f4*)(colsum + col);
    f4 o;
    o.x = (w & 1u) ? (l4.x + s4.x - cv.x) : 0.f;
    o.y = (w & 2u) ? (l4.y + s4.y - cv.y) : 0.f;
    o.z = (w & 4u) ? (l4.z + s4.z - cv.z) : 0.f;
    o.w = (w & 8u) ? (l4.w + s4.w - cv.w) : 0.f;
    *(f4*)(v2c + e) = o;
}

// ---- posterior + hard decision + iteration count ----
__global__ void k_final(const float* __restrict__ llr, const float* __restrict__ colsum,
                        int* __restrict__ dec, float* __restrict__ post,
                        int* __restrict__ iters, int T) {
    int n = blockIdx.x * blockDim.x + threadIdx.x;
    if (n < Ndim) {
        float p = llr[n] + colsum[n];
        post[n] = p;
        dec[n]  = (p < 0.f) ? 1 : 0;
        if (n == 0) *iters = T;
    }
}

extern "C" void kernel_launch(void* const* d_in, const int* in_sizes, int n_in,
                              void* d_out, int out_size, void* d_ws, size_t ws_size,
                              hipStream_t stream) {
    const float* llr  = (const float*)d_in[0];
    const float* beta = (const float*)d_in[1];
    const float* H    = (const float*)d_in[2];
    const size_t MN = (size_t)Mdim * Ndim;
    const int T = (int)(in_sizes[1] / (long long)MN);

    char* ws = (char*)d_ws;
    float*    v2c    = (float*)ws;                                  // 32 MB
    float*    c2v    = (float*)(ws + MN * 4);                       // 32 MB
    float*    colsum = (float*)(ws + 2 * MN * 4);                   // 16 KB
    uint32_t* bits   = (uint32_t*)(ws + 2 * MN * 4 + Ndim * 4);     // 1 MB

    k_pack_mask<<<(int)(MN / 256), 256, 0, stream>>>(H, bits);
    k_init_v2c<<<(int)(MN / 1024), 256, 0, stream>>>(bits, llr, v2c);

    for (int t = 0; t < T; ++t) {
        k_zero<<<(Ndim + 255) / 256, 256, 0, stream>>>(colsum, Ndim);
        k_check<<<Mdim / ROWS_PER_BLOCK, 128, 0, stream>>>(
            v2c, beta + (size_t)t * MN, bits, c2v);
        k_colsum<<<(Ndim / 16) * 8 / 8, 256, 0, stream>>>(c2v, colsum);
        if (t + 1 < T)
            k_var<<<(int)(MN / 1024), 256, 0, stream>>>(c2v, bits, llr, colsum, v2c);
    }

    int*   dec   = (int*)d_out;
    float* post  = (float*)d_out + Ndim;
    int*   iters = (int*)d_out + 2 * Ndim;
    k_final<<<(Ndim + 255) / 256, 256, 0, stream>>>(llr, colsum, dec, post, iters, T);
}